// GATLayer_4252017623408
// MI455X (gfx1250) — compile-verified
//
#include <hip/hip_runtime.h>
#include <hip/hip_bf16.h>

typedef float v2f __attribute__((ext_vector_type(2)));
typedef float v8f __attribute__((ext_vector_type(8)));

#define DFEAT 128
#define NHEAD 4
#define DHEAD 32

// ---------------------------------------------------------------------------
// C[M x 128] = (A[M x 128] @ W[128 x 128] + bias[128]) * scale
// One wave per 16x16 output tile, 8 waves per block -> 16x128 row slab.
// Uses V_WMMA_F32_16X16X4_F32 (CDNA5 f32 matrix path), K-loop of 32 steps.
// ---------------------------------------------------------------------------
__global__ void __launch_bounds__(256) gemm128_wmma(
    const float* __restrict__ A, const float* __restrict__ W,
    const float* __restrict__ bias, float* __restrict__ C, float scale)
{
    const int rowBase = blockIdx.x * 16;
    const int wave    = threadIdx.x >> 5;
    const int lane    = threadIdx.x & 31;
    const int colBase = wave * 16;
    const int r       = lane & 15;     // M row (A) / N col (B) within tile
    const int halfSel = lane >> 4;     // selects K pair {0,1} vs {2,3}

    v8f acc = {0.f,0.f,0.f,0.f,0.f,0.f,0.f,0.f};
    const float* __restrict__ Arow = A + (size_t)(rowBase + r) * DFEAT;

    #pragma unroll 4
    for (int kb = 0; kb < DFEAT; kb += 4) {
        const int ka = kb + 2 * halfSel;
        v2f a, b;
        a.x = Arow[ka];
        a.y = Arow[ka + 1];
        b.x = W[(size_t)ka       * DFEAT + colBase + r];
        b.y = W[(size_t)(ka + 1) * DFEAT + colBase + r];
        acc = __builtin_amdgcn_wmma_f32_16x16x4_f32(
                  /*neg_a=*/false, a, /*neg_b=*/false, b,
                  /*c_mod=*/(short)0, acc, /*reuse_a=*/false, /*reuse_b=*/false);
    }

    const int col = colBase + r;
    const float bv = bias[col];
    #pragma unroll
    for (int v = 0; v < 8; ++v) {
        const int rowOut = rowBase + halfSel * 8 + v;   // ISA C/D layout
        C[(size_t)rowOut * DFEAT + col] = (acc[v] + bv) * scale;
    }
}

// ---------------------------------------------------------------------------
// Zero/initialize accumulation state every call (graph-replay safe).
// smaxU uses a monotonic uint encoding of float; 0 is below every encoding.
// ---------------------------------------------------------------------------
__global__ void init_kernel(float* __restrict__ agg, int nAgg,
                            unsigned* __restrict__ smaxU,
                            float* __restrict__ ssum, int nNH,
                            float* __restrict__ cnt, int nN)
{
    const int i = blockIdx.x * blockDim.x + threadIdx.x;
    if (i < nAgg) agg[i] = 0.f;
    if (i < nNH) { smaxU[i] = 0u; ssum[i] = 0.f; }
    if (i < nN)  cnt[i] = 0.f;
}

__device__ __forceinline__ unsigned encodeF(float f) {
    unsigned u = __float_as_uint(f);
    return (u & 0x80000000u) ? ~u : (u | 0x80000000u);
}
__device__ __forceinline__ float decodeF(unsigned u) {
    return (u & 0x80000000u) ? __uint_as_float(u ^ 0x80000000u)
                             : __uint_as_float(~u);
}

// ---------------------------------------------------------------------------
// One wave per edge: 4 head-dots of width 32, wave32 xor-shuffle reduction,
// atomic segment-max over src, and per-src edge count.
// ---------------------------------------------------------------------------
__global__ void __launch_bounds__(256) score_kernel(
    const float* __restrict__ Q, const float* __restrict__ K,
    const int* __restrict__ edge, float* __restrict__ scoreBuf,
    unsigned* __restrict__ smaxU, float* __restrict__ cnt, int E)
{
    const int wave = threadIdx.x >> 5;
    const int lane = threadIdx.x & 31;
    const int e = blockIdx.x * 8 + wave;
    if (e >= E) return;
    const int src = edge[e];
    const int dst = edge[E + e];
    const float* __restrict__ q = Q + (size_t)src * DFEAT;
    const float* __restrict__ k = K + (size_t)dst * DFEAT;

    #pragma unroll
    for (int h = 0; h < NHEAD; ++h) {
        float p = q[h * DHEAD + lane] * k[h * DHEAD + lane];
        #pragma unroll
        for (int off = 16; off > 0; off >>= 1)
            p += __shfl_xor(p, off, 32);
        if (lane == 0) {
            scoreBuf[(size_t)e * NHEAD + h] = p;
            atomicMax(&smaxU[src * NHEAD + h], encodeF(p));
        }
    }
    if (lane == 0) atomicAdd(&cnt[src], 1.0f);
}

// ---------------------------------------------------------------------------
// exp(score - segmax) and segment sum. One thread per (edge, head).
// ---------------------------------------------------------------------------
__global__ void expsum_kernel(const int* __restrict__ edge,
                              const unsigned* __restrict__ smaxU,
                              float* __restrict__ scoreBuf,
                              float* __restrict__ ssum, int E)
{
    const int i = blockIdx.x * blockDim.x + threadIdx.x;
    if (i >= E * NHEAD) return;
    const int e = i >> 2;
    const int h = i & 3;
    const int src = edge[e];
    const float m = decodeF(smaxU[src * NHEAD + h]);
    const float ev = __expf(scoreBuf[i] - m);
    scoreBuf[i] = ev;
    atomicAdd(&ssum[src * NHEAD + h], ev);
}

// ---------------------------------------------------------------------------
// alpha = e/ssum[src] * cnt[src]; agg[dst] += M[src] * alpha. Wave per edge,
// each lane handles one of the 32 channels of each head.
// ---------------------------------------------------------------------------
__global__ void __launch_bounds__(256) scatter_kernel(
    const int* __restrict__ edge, const float* __restrict__ Mbuf,
    const float* __restrict__ scoreBuf, const float* __restrict__ ssum,
    const float* __restrict__ cnt, float* __restrict__ agg, int E)
{
    const int wave = threadIdx.x >> 5;
    const int lane = threadIdx.x & 31;
    const int e = blockIdx.x * 8 + wave;
    if (e >= E) return;
    const int src = edge[e];
    const int dst = edge[E + e];
    const float c = cnt[src];

    #pragma unroll
    for (int h = 0; h < NHEAD; ++h) {
        const float alpha = scoreBuf[(size_t)e * NHEAD + h]
                          / ssum[src * NHEAD + h] * c;
        const float mv = Mbuf[(size_t)src * DFEAT + h * DHEAD + lane];
        atomicAdd(&agg[(size_t)dst * DFEAT + h * DHEAD + lane], mv * alpha);
    }
}

// ---------------------------------------------------------------------------
// BatchNorm batch statistics: one block per feature column.
// ---------------------------------------------------------------------------
__global__ void __launch_bounds__(256) bnstats_kernel(
    const float* __restrict__ H, int N,
    float* __restrict__ meanv, float* __restrict__ rsig)
{
    const int c = blockIdx.x;
    float s = 0.f, s2 = 0.f;
    for (int r = threadIdx.x; r < N; r += 256) {
        const float v = H[(size_t)r * DFEAT + c];
        s += v; s2 += v * v;
    }
    __shared__ float sh[256], sh2[256];
    sh[threadIdx.x] = s; sh2[threadIdx.x] = s2;
    __syncthreads();
    for (int st = 128; st > 0; st >>= 1) {
        if (threadIdx.x < st) {
            sh[threadIdx.x]  += sh[threadIdx.x + st];
            sh2[threadIdx.x] += sh2[threadIdx.x + st];
        }
        __syncthreads();
    }
    if (threadIdx.x == 0) {
        const float mean = sh[0] / (float)N;
        const float var  = sh2[0] / (float)N - mean * mean;  // biased, like torch BN
        meanv[c] = mean;
        rsig[c]  = rsqrtf(var + 1e-5f);
    }
}

__global__ void bnrelu_kernel(float* __restrict__ H,
                              const float* __restrict__ meanv,
                              const float* __restrict__ rsig,
                              const float* __restrict__ gamma,
                              const float* __restrict__ beta, int total)
{
    const int i = blockIdx.x * blockDim.x + threadIdx.x;
    if (i >= total) return;
    const int c = i & (DFEAT - 1);
    const float v = (H[i] - meanv[c]) * rsig[c] * gamma[c] + beta[c];
    H[i] = v > 0.f ? v : 0.f;
}

// ---------------------------------------------------------------------------
extern "C" void kernel_launch(void* const* d_in, const int* in_sizes, int n_in,
                              void* d_out, int out_size, void* d_ws, size_t ws_size,
                              hipStream_t stream) {
    const float* x     = (const float*)d_in[0];
    const int*   edge  = (const int*)  d_in[1];   // [2, E]: row0=src, row1=dst
    const float* Wk    = (const float*)d_in[2];
    const float* bk    = (const float*)d_in[3];
    const float* Wm    = (const float*)d_in[4];
    const float* bm    = (const float*)d_in[5];
    const float* Wq    = (const float*)d_in[6];
    const float* bq    = (const float*)d_in[7];
    const float* W1    = (const float*)d_in[8];
    const float* b1    = (const float*)d_in[9];
    const float* gamma = (const float*)d_in[10];
    const float* beta  = (const float*)d_in[11];
    const float* W2    = (const float*)d_in[12];
    const float* b2    = (const float*)d_in[13];

    const int N = in_sizes[0] / DFEAT;   // 50000 (multiple of 16)
    const int E = in_sizes[1] / 2;       // 800000

    // Workspace layout (floats). hbuf aliases Qbuf (Q dead after score pass).
    float* ws      = (float*)d_ws;
    float* Kbuf    = ws;
    float* Mbuf    = Kbuf + (size_t)N * DFEAT;
    float* Qbuf    = Mbuf + (size_t)N * DFEAT;
    float* agg     = Qbuf + (size_t)N * DFEAT;
    unsigned* smaxU = (unsigned*)(agg + (size_t)N * DFEAT);
    float* ssum    = (float*)(smaxU + (size_t)N * NHEAD);
    float* cnt     = ssum + (size_t)N * NHEAD;
    float* meanv   = cnt + N;
    float* rsig    = meanv + DFEAT;
    float* scoreBuf = rsig + DFEAT;
    float* hbuf    = Qbuf;               // reuse

    const int total = N * DFEAT;
    const float qscale = 0.17677669529663687f;   // 1/sqrt(32)

    init_kernel<<<(total + 255) / 256, 256, 0, stream>>>(
        agg, total, smaxU, ssum, N * NHEAD, cnt, N);

    // Per-node projections (16x cheaper than per-edge) on the f32 WMMA path.
    gemm128_wmma<<<N / 16, 256, 0, stream>>>(x, Wk, bk, Kbuf, 1.0f);
    gemm128_wmma<<<N / 16, 256, 0, stream>>>(x, Wm, bm, Mbuf, 1.0f);
    gemm128_wmma<<<N / 16, 256, 0, stream>>>(x, Wq, bq, Qbuf, qscale);

    score_kernel<<<(E + 7) / 8, 256, 0, stream>>>(
        Qbuf, Kbuf, edge, scoreBuf, smaxU, cnt, E);
    expsum_kernel<<<(E * NHEAD + 255) / 256, 256, 0, stream>>>(
        edge, smaxU, scoreBuf, ssum, E);
    scatter_kernel<<<(E + 7) / 8, 256, 0, stream>>>(
        edge, Mbuf, scoreBuf, ssum, cnt, agg, E);

    // MLP: Linear -> BN(batch stats) -> ReLU -> Linear
    gemm128_wmma<<<N / 16, 256, 0, stream>>>(agg, W1, b1, hbuf, 1.0f);
    bnstats_kernel<<<DFEAT, 256, 0, stream>>>(hbuf, N, meanv, rsig);
    bnrelu_kernel<<<(total + 255) / 256, 256, 0, stream>>>(
        hbuf, meanv, rsig, gamma, beta, total);
    gemm128_wmma<<<N / 16, 256, 0, stream>>>(hbuf, W2, b2, (float*)d_out, 1.0f);
}